// ButterflyModule_2061584302533
// MI455X (gfx1250) — compile-verified
//
#include <hip/hip_runtime.h>
#include <math.h>

// CDNA5 / gfx1250 butterfly network as one f32 WMMA GEMM.
//
// Roofline: 512 MB of HBM traffic @ 23.3 TB/s => ~22 us floor; compute is
// ~17 GFLOP even in dense-GEMM form, so this is purely bandwidth-bound.
// The 8 butterfly layers all rotate the same permuted feature pairs, so they
// compose into a single 128x128 rotation matrix M (2x2-block sparse, 64 KB)
// with out_row = M * in_row.  We run out = in * M^T with
// V_WMMA_F32_16X16X4_F32, keeping full fp32 precision (zero entries of M
// contribute exact zeros to the WMMA dot products).

typedef __attribute__((ext_vector_type(2))) float v2f;
typedef __attribute__((ext_vector_type(8))) float v8f;

#define NFEAT        128
#define NPAIR        64
#define NLAYER       8
#define LDS_STRIDE   132   // 128 + 4 pad: row stride = 4 banks mod 64 =>
                           // conflict-free ds_load_b64 B-fragment fetches
#define ROWS_PER_BLK 128
#define THREADS      256   // 8 wave32 waves

// ---------------------------------------------------------------------------
// Kernel 1: fold the 8 layers into one 128x128 rotation matrix M (row-major,
// M[out_feat][in_feat]).  One block, one thread per pair.  Rotation
// composition is done as a 2x2 matrix product chain, numerically matching the
// reference's sequential application.
// ---------------------------------------------------------------------------
__global__ __launch_bounds__(NPAIR) void build_rotation_matrix(
    const float* __restrict__ angles,   // [NLAYER][NPAIR]
    const int*   __restrict__ idx_in,   // [NFEAT] permutation
    const int*   __restrict__ idx_out,  // [NFEAT] permutation
    float*       __restrict__ M)        // [NFEAT][NFEAT] in workspace
{
    const int j = threadIdx.x;          // pair id 0..63

    // zero M cooperatively: 64 threads x 256 contiguous floats = 16384
    for (int t = 0; t < NFEAT * NFEAT / NPAIR; ++t)
        M[j * (NFEAT * NFEAT / NPAIR) + t] = 0.0f;
    __syncthreads();

    float c = 1.0f, s = 0.0f;
    #pragma unroll
    for (int l = 0; l < NLAYER; ++l) {
        const float a  = angles[l * NPAIR + j];
        const float cl = cosf(a);
        const float sl = sinf(a);
        const float cn = c * cl - s * sl;   // compose R(a) * R(acc)
        const float sn = s * cl + c * sl;
        c = cn; s = sn;
    }

    const int pi0 = idx_in [2 * j], pi1 = idx_in [2 * j + 1];
    const int po0 = idx_out[2 * j], po1 = idx_out[2 * j + 1];
    M[po0 * NFEAT + pi0] =  c;
    M[po0 * NFEAT + pi1] = -s;
    M[po1 * NFEAT + pi0] =  s;
    M[po1 * NFEAT + pi1] =  c;
}

// ---------------------------------------------------------------------------
// Kernel 2: out[b, i] = sum_k in[b, k] * M[i, k]  via V_WMMA_F32_16X16X4_F32.
// Block = 8 waves; wave w owns batch rows [blk*128 + 16w, +16) and all 8
// 16-wide feature tiles (8 x v8f accumulators).  M staged once per block into
// padded LDS.
//
// Fragment layouts (CDNA5 ISA):
//   A 16x4 f32, 2 VGPRs:  lane<16 -> K = k0,k0+1 ; lane>=16 -> K = k0+2,k0+3
//   B 4x16 f32, 2 VGPRs:  row k in VGPR k%2, lane-half k/2, N = lane%16
//   C/D 16x16 f32, 8 VGPRs: VGPR g -> M = g (+8 for upper half), N = lane%16
// ---------------------------------------------------------------------------
__global__ __launch_bounds__(THREADS) void butterfly_wmma(
    const float* __restrict__ in,
    const float* __restrict__ M,
    float*       __restrict__ out)
{
    __shared__ float Ms[NFEAT * LDS_STRIDE];   // 66 KB (WGP LDS = 320 KB)

    const int t = threadIdx.x;

    // Stage M: 4096 float4, coalesced global reads, 16B-aligned LDS stores.
    for (int idx = t; idx < NFEAT * (NFEAT / 4); idx += THREADS) {
        const int    i  = idx >> 5;          // M row (feature i)
        const int    k4 = (idx & 31) << 2;   // k within row
        const float4 v  = ((const float4*)M)[idx];
        *(float4*)&Ms[i * LDS_STRIDE + k4] = v;
    }
    __syncthreads();

    const int wave = t >> 5;
    const int lane = t & 31;
    const int half = lane >> 4;   // 0: lanes 0-15, 1: lanes 16-31
    const int l16  = lane & 15;

    const long  row0  = (long)blockIdx.x * ROWS_PER_BLK + wave * 16;
    const float* Abase = in + (row0 + l16) * NFEAT + half * 2;

    v8f acc[8] = {};

    for (int k0 = 0; k0 < NFEAT; k0 += 4) {
        // A fragment: lane holds in[row0+l16, k0+2*half .. +1] (8B contiguous)
        const v2f a = *(const v2f*)(Abase + k0);
        #pragma unroll
        for (int n = 0; n < 8; ++n) {
            // B fragment element v = B[k0+2*half+v][col] = M[col][k0+2*half+v]
            const v2f b = *(const v2f*)
                &Ms[(n * 16 + l16) * LDS_STRIDE + k0 + half * 2];
            acc[n] = __builtin_amdgcn_wmma_f32_16x16x4_f32(
                /*neg_a=*/false, a, /*neg_b=*/false, b,
                /*c_mod=*/(short)0, acc[n],
                /*reuse_a=*/false, /*reuse_b=*/false);
        }
    }

    // D store: lane lane, VGPR g -> out[row0 + 8*half + g, 16n + l16]
    float* Obase = out + (row0 + half * 8) * NFEAT + l16;
    #pragma unroll
    for (int n = 0; n < 8; ++n) {
        #pragma unroll
        for (int g = 0; g < 8; ++g) {
            Obase[(long)g * NFEAT + n * 16] = acc[n][g];
        }
    }
}

// ---------------------------------------------------------------------------
extern "C" void kernel_launch(void* const* d_in, const int* in_sizes, int n_in,
                              void* d_out, int out_size, void* d_ws, size_t ws_size,
                              hipStream_t stream) {
    const float* data    = (const float*)d_in[0];   // [BATCH][128] f32
    const float* angles  = (const float*)d_in[1];   // [8][64] f32
    const int*   idx_in  = (const int*)  d_in[2];   // [128] i32
    const int*   idx_out = (const int*)  d_in[3];   // [128] i32
    float*       out     = (float*)d_out;
    float*       M       = (float*)d_ws;            // 64 KB scratch

    const long batch  = (long)in_sizes[0] / NFEAT;  // 524288
    const int  blocks = (int)(batch / ROWS_PER_BLK);// 4096

    build_rotation_matrix<<<1, NPAIR, 0, stream>>>(angles, idx_in, idx_out, M);
    butterfly_wmma<<<blocks, THREADS, 0, stream>>>(data, M, out);
}